// MpmP2G_34041910788794
// MI455X (gfx1250) — compile-verified
//
#include <hip/hip_runtime.h>

// Grid constants (from reference): padded grid 130^3, inner 128^3, DX = 1/128
#define NPAD   130
#define HWROW  (130 * 130)            // H*W
#define NCELL  (130 * 130 * 130)      // 2,197,000 cells
#define NOUT   (NCELL * 3)            // 6,591,000 floats in d_out
#define N_OUT4 (NOUT / 4)             // 1,647,750  (exact)
#define N_W4   (NCELL / 4)            // 549,250    (exact)

#if defined(__has_builtin)
#if __has_builtin(__builtin_amdgcn_global_load_async_to_lds_b32)
#define P2G_HAVE_ASYNC_LDS 1
#endif
#endif

typedef __attribute__((address_space(1))) int gint;   // global (AS1)
typedef __attribute__((address_space(3))) int lint;   // LDS (AS3)

// ---------------------------------------------------------------------------
// Kernel 1: zero the accumulators (d_out = grid_value, d_ws = grid_weight).
// These lines must stay resident in L2 (the atomics will hammer them), so use
// regular-temporal stores.
// ---------------------------------------------------------------------------
__global__ __launch_bounds__(256) void p2g_init(float4* __restrict__ out4,
                                                float4* __restrict__ w4) {
    int t = blockIdx.x * blockDim.x + threadIdx.x;
    float4 z = make_float4(0.f, 0.f, 0.f, 0.f);
    if (t < N_OUT4) out4[t] = z;
    if (t < N_W4)   w4[t]   = z;
}

// ---------------------------------------------------------------------------
// Kernel 2: P2G scatter. One particle per lane. The AoS (N,3) particle tile
// for the block (256 particles, 2 x 3 KB) is staged into LDS:
//  - preferred: gfx1250 async global->LDS copy (GLOBAL_LOAD_ASYNC_TO_LDS_B32,
//    ASYNCcnt-tracked, no VGPR round trip), completed with s_wait_asynccnt 0
//    before the workgroup barrier;
//  - fallback: coalesced non-temporal loads + ds stores.
// Stride-3 LDS reads are bank-conflict-free (3 coprime with 64 banks).
// 27-cell quadratic B-spline stencil fully unrolled; accumulation via native
// f32 global atomics (global_atomic_add_f32 -> L2 atomic units).
// ---------------------------------------------------------------------------
__global__ __launch_bounds__(256) void p2g_scatter(const float* __restrict__ inp,
                                                   const float* __restrict__ pos,
                                                   float* __restrict__ out,
                                                   float* __restrict__ wsum,
                                                   int n) {
    __shared__ float sp[768];   // 256 particles * 3 (pos)
    __shared__ float sv[768];   // 256 particles * 3 (inp)

    const int tid   = threadIdx.x;
    const int pbase = blockIdx.x * 256;
    const int gbase = pbase * 3;
    const int avail = n * 3 - gbase;   // valid floats in this block's window

#if defined(P2G_HAVE_ASYNC_LDS)
#pragma unroll
    for (int k = 0; k < 3; ++k) {
        int off = tid + k * 256;
        if (off < avail) {
            __builtin_amdgcn_global_load_async_to_lds_b32(
                (gint*)(pos + gbase + off), (lint*)&sp[off], 0, 0);
            __builtin_amdgcn_global_load_async_to_lds_b32(
                (gint*)(inp + gbase + off), (lint*)&sv[off], 0, 0);
        }
    }
    asm volatile("s_wait_asynccnt 0x0" ::: "memory");
#else
#pragma unroll
    for (int k = 0; k < 3; ++k) {
        int off = tid + k * 256;
        if (off < avail) {
            sp[off] = __builtin_nontemporal_load(pos + gbase + off);
            sv[off] = __builtin_nontemporal_load(inp + gbase + off);
        }
    }
#endif
    __syncthreads();

    const int i = pbase + tid;
    if (i >= n) return;

    const float px = sp[3 * tid + 0], py = sp[3 * tid + 1], pz = sp[3 * tid + 2];
    const float vx = sv[3 * tid + 0], vy = sv[3 * tid + 1], vz = sv[3 * tid + 2];

    // normalized grid coords: (pos - 0) / (1/128) = pos * 128, in [0,128)
    const float nx = px * 128.0f, ny = py * 128.0f, nz = pz * 128.0f;
    const int gx = (int)nx, gy = (int)ny, gz = (int)nz;
    const float lx = nx - (float)gx, ly = ny - (float)gy, lz = nz - (float)gz;

    // quadratic B-spline weights per axis
    float wx[3], wy[3], wz[3];
    wx[0] = 0.5f * (1.0f - lx) * (1.0f - lx);
    wx[1] = 0.75f - (0.5f - lx) * (0.5f - lx);
    wx[2] = 0.5f * lx * lx;
    wy[0] = 0.5f * (1.0f - ly) * (1.0f - ly);
    wy[1] = 0.75f - (0.5f - ly) * (0.5f - ly);
    wy[2] = 0.5f * ly * ly;
    wz[0] = 0.5f * (1.0f - lz) * (1.0f - lz);
    wz[1] = 0.75f - (0.5f - lz) * (0.5f - lz);
    wz[2] = 0.5f * lz * lz;

    const int base = (gx * NPAD + gy) * NPAD + gz;

#pragma unroll
    for (int ox = 0; ox < 3; ++ox) {
        const int bx = base + ox * HWROW;
        const float wxo = wx[ox];
#pragma unroll
        for (int oy = 0; oy < 3; ++oy) {
            const float wxy = wxo * wy[oy];
            const int bxy = bx + oy * NPAD;
#pragma unroll
            for (int oz = 0; oz < 3; ++oz) {
                const float w = wxy * wz[oz];
                const int idx = bxy + oz;
                float* o = out + 3 * idx;
                unsafeAtomicAdd(o + 0, vx * w);
                unsafeAtomicAdd(o + 1, vy * w);
                unsafeAtomicAdd(o + 2, vz * w);
                unsafeAtomicAdd(wsum + idx, w);
            }
        }
    }
}

// ---------------------------------------------------------------------------
// Kernel 3: normalize in place: out[c] = value[c] / weight (weight==0 -> keep,
// value is 0 there anyway). True IEEE division to match reference rounding.
// ---------------------------------------------------------------------------
__global__ __launch_bounds__(256) void p2g_norm(float* __restrict__ out,
                                                const float* __restrict__ wsum) {
    int i = blockIdx.x * blockDim.x + threadIdx.x;
    if (i >= NCELL) return;
    float w = wsum[i];
    if (w != 0.0f) {
        out[3 * i + 0] = out[3 * i + 0] / w;
        out[3 * i + 1] = out[3 * i + 1] / w;
        out[3 * i + 2] = out[3 * i + 2] / w;
    }
}

// ---------------------------------------------------------------------------
extern "C" void kernel_launch(void* const* d_in, const int* in_sizes, int n_in,
                              void* d_out, int out_size, void* d_ws, size_t ws_size,
                              hipStream_t stream) {
    const float* inp = (const float*)d_in[0];   // (N,3) particle values
    const float* pos = (const float*)d_in[1];   // (N,3) particle positions
    float* out  = (float*)d_out;                // (130,130,130,3) accumulator / result
    float* wsum = (float*)d_ws;                 // (130^3) weight accumulator

    const int n = in_sizes[0] / 3;              // particle count

    const int initBlocks = (N_OUT4 + 255) / 256;
    p2g_init<<<initBlocks, 256, 0, stream>>>((float4*)d_out, (float4*)d_ws);

    const int scatBlocks = (n + 255) / 256;
    p2g_scatter<<<scatBlocks, 256, 0, stream>>>(inp, pos, out, wsum, n);

    const int normBlocks = (NCELL + 255) / 256;
    p2g_norm<<<normBlocks, 256, 0, stream>>>(out, wsum);
}